// BatchedBikeModel_35338990911884
// MI455X (gfx1250) — compile-verified
//
#include <hip/hip_runtime.h>
#include <hip/hip_bf16.h>

// Bicycle-model batched scan, B=65536, T=256, state=5 (x,y,h,v,hd).
// One thread per batch element (recurrence is sequential in T).
// Output (B,T,5) staged in LDS per 8-step chunk, then scattered with the
// CDNA5 async store engine (global_store_async_from_lds_b128, NT hint so the
// 320MB write-once stream doesn't thrash the 192MB L2) while ALUs continue.

#define DT_F      0.1f
#define TT        256
#define CHUNK     8
#define NCHUNK    (TT / CHUNK)          // 32
#define THREADS   128
#define ROW_F     44                    // CHUNK*5 = 40 floats, padded to 44 (176B, 16B-aligned)
#define VEC_PER_T (CHUNK * 5 / 4)       // 10 float4 per thread per chunk

// atan(z) for |z| <= ~0.32 (clamp args are x/100 with x bounded by model
// dynamics: yaw-rate state in [-1.2,1.2], speed state in [-1,31]).
// Degree-9 odd Taylor poly; rel err < 1e-5, far cheaper than libm atanf.
__device__ __forceinline__ float atan_small(float z) {
    float z2 = z * z;
    float p = fmaf(z2, 0.11111111f, -0.14285714f);
    p = fmaf(z2, p, 0.2f);
    p = fmaf(z2, p, -0.33333333f);
    p = fmaf(z2, p, 1.0f);
    return z * p;
}

__device__ __forceinline__ float soft_clamp(float x, float lo, float hi) {
    // lo + (hi-lo)*(0.5 + atan(x/100)/pi)
    float t = fmaf(atan_small(x * 0.01f), 0.318309886183790672f, 0.5f);
    return fmaf(hi - lo, t, lo);
}

__global__ __launch_bounds__(THREADS)
void bike_scan_kernel(const float* __restrict__ init_state,
                      const float* __restrict__ accel,
                      const float* __restrict__ steer,
                      const float* __restrict__ veh_len,
                      float* __restrict__ out)
{
    __shared__ __align__(16) float stage[2][THREADS][ROW_F];

    const int tid      = threadIdx.x;
    const int lane     = tid & 31;
    const int waveBase = tid & ~31;
    const int b        = blockIdx.x * THREADS + tid;

    // ---- initial state ----
    const float* is = init_state + (size_t)b * 5;
    float x  = is[0];
    float y  = is[1];
    float h  = is[2];
    float v  = is[3];
    float hd = is[4];
    const float invL = 1.0f / veh_len[b];
    const float kYaw = DT_F * invL;

    const float4* ap4 = (const float4*)(accel + (size_t)b * TT);
    const float4* sp4 = (const float4*)(steer + (size_t)b * TT);

    float carry_a = 0.0f, carry_s = 0.0f;

    for (int c = 0; c < NCHUNK; ++c) {
        // Double-buffer guard: each chunk issues VEC_PER_T async-store insts
        // (wave-wide). Waiting to <=10 guarantees the buffer we are about to
        // overwrite (used by chunk c-2) has been fully read by the engine.
        asm volatile("s_wait_asynccnt 10" ::: "memory");

        const int buf = c & 1;
        float* st = &stage[buf][tid][0];

        // prefetch next chunk of inputs into the cache hierarchy
        const int cn = (c < NCHUNK - 1) ? (c + 1) : c;
        __builtin_prefetch((const void*)(ap4 + (size_t)cn * 2), 0, 3);
        __builtin_prefetch((const void*)(sp4 + (size_t)cn * 2), 0, 3);

        // ---- load this chunk's inputs (2 float4 per stream) ----
        float av[CHUNK], sv[CHUNK];
        {
            float4 q;
            q = ap4[c * 2 + 0]; av[0] = q.x; av[1] = q.y; av[2] = q.z; av[3] = q.w;
            q = ap4[c * 2 + 1]; av[4] = q.x; av[5] = q.y; av[6] = q.z; av[7] = q.w;
            q = sp4[c * 2 + 0]; sv[0] = q.x; sv[1] = q.y; sv[2] = q.z; sv[3] = q.w;
            q = sp4[c * 2 + 1]; sv[4] = q.x; sv[5] = q.y; sv[6] = q.z; sv[7] = q.w;
        }

        // ---- CHUNK sequential steps; stage states into LDS ----
        // Output index n = c*CHUNK + i. n==0 (c==0,i==0) is the initial
        // state: skip the step but stage the loaded state.
        #pragma unroll
        for (int i = 0; i < CHUNK; ++i) {
            const float ta = (i == 0) ? carry_a : av[i - 1];
            const float ts = (i == 0) ? carry_s : sv[i - 1];
            if (i != 0 || c != 0) {
                float nhd = soft_clamp(fmaf(ts, DT_F, hd), -1.0f, 1.0f);
                float nh  = fmaf(kYaw * fabsf(v), nhd, h);
                float ns  = soft_clamp(fmaf(ta, DT_F, v), 0.0f, 30.0f);
                float sn, cs;
                __sincosf(nh, &sn, &cs);
                y = fmaf(ns * sn, DT_F, y);
                x = fmaf(ns * cs, DT_F, x);
                h = nh; v = ns; hd = nhd;
            }
            st[i * 5 + 0] = x;
            st[i * 5 + 1] = y;
            st[i * 5 + 2] = h;
            st[i * 5 + 3] = v;
            st[i * 5 + 4] = hd;
        }
        carry_a = av[CHUNK - 1];
        carry_s = sv[CHUNK - 1];

        // DS writes must be committed before the async engine reads LDS.
        asm volatile("s_wait_dscnt 0" ::: "memory");

        // ---- wave-cooperative async scatter: 32 rows x 10 float4 ----
        // Flat mapping f = it*32 + lane; row = f/10, vec = f%10: each
        // async-store instruction covers a contiguous run of output rows
        // (good per-instruction coalescing), and each unrolled iteration's
        // addressing derives independently from `lane` (ILP after the wait).
        const unsigned ldsWave = (unsigned)(size_t)&stage[buf][waveBase][0];
        const size_t   gRow0   = (size_t)(blockIdx.x * THREADS + waveBase);
        const float*   gChunk  = out + gRow0 * (size_t)(TT * 5)
                                     + (size_t)(c * CHUNK * 5);
        #pragma unroll
        for (int it = 0; it < VEC_PER_T; ++it) {
            unsigned f   = (unsigned)(it * 32 + lane);
            unsigned bw  = f / 10u;              // row within wave: 0..31
            unsigned vec = f - bw * 10u;         // float4 within row: 0..9
            unsigned lofs = ldsWave + bw * (ROW_F * 4u) + vec * 16u;
            unsigned long long ga = (unsigned long long)(size_t)(gChunk
                + (size_t)bw * (size_t)(TT * 5)   // batch row (5120 B)
                + (size_t)(vec * 4));             // float4 within 160B burst
            // Non-temporal: output is a 320MB write-once stream (> 192MB L2);
            // keep it from evicting the reused input-stream cachelines.
            asm volatile("global_store_async_from_lds_b128 %0, %1, off th:TH_STORE_NT"
                         :: "v"(ga), "v"(lofs) : "memory");
        }
        // no barrier needed: staging regions are per-wave private
    }
    // S_ENDPGM performs an implicit wait-idle, draining remaining async stores.
}

extern "C" void kernel_launch(void* const* d_in, const int* in_sizes, int n_in,
                              void* d_out, int out_size, void* d_ws, size_t ws_size,
                              hipStream_t stream) {
    const float* init = (const float*)d_in[0];   // (B,5)
    const float* acc  = (const float*)d_in[1];   // (B,T)
    const float* str  = (const float*)d_in[2];   // (B,T)
    const float* vlen = (const float*)d_in[3];   // (B,)
    float* out = (float*)d_out;                  // (B,T,5)

    const int B = in_sizes[3];
    dim3 grid(B / THREADS);
    dim3 block(THREADS);
    hipLaunchKernelGGL(bike_scan_kernel, grid, block, 0, stream,
                       init, acc, str, vlen, out);
}